// BlurPool3d_35424890257760
// MI455X (gfx1250) — compile-verified
//
#include <hip/hip_runtime.h>

// BlurPool3d (depthwise 3x3x3 binomial blur, stride 2, replicate pad), FP32.
// Bandwidth-bound (AI ~1.5 flop/byte) -> optimize data movement with the
// CDNA5 async global->LDS DMA path and a 5-slot rotating plane buffer so each
// input plane is fetched from HBM once per (n,c,oh-tile) block.

#define NB    4
#define CH    64
#define IDm   32
#define IHm   112
#define IWm   112
#define ODm   16
#define OHm   56
#define OWm   56
#define OHT   8                 // output rows per tile
#define NROWS 17                // input rows staged per plane slot (2*OHT+1)
#define ROWF  112               // floats per staged row (full W)
#define PLANEF (NROWS * ROWF)   // 1904 floats per plane slot
#define NSLOT 5                 // rotating plane slots (3 in use + 2 loading)
#define NVEC  (NROWS * (ROWF / 4))  // 476 x 16B chunks per plane

#ifndef __has_builtin
#define __has_builtin(x) 0
#endif

typedef int v4i __attribute__((vector_size(16)));
typedef __attribute__((address_space(1))) v4i Gv4;   // global int4
typedef __attribute__((address_space(3))) v4i Lv4;   // LDS int4
typedef __attribute__((address_space(3))) void LdsVoid;

__device__ __forceinline__ void async_copy_b128(const float* g, float* l) {
#if __has_builtin(__builtin_amdgcn_global_load_async_to_lds_b128)
  __builtin_amdgcn_global_load_async_to_lds_b128((Gv4*)g, (Lv4*)l, 0, 0);
#else
  unsigned laddr = (unsigned)(size_t)(LdsVoid*)l;
  asm volatile("global_load_async_to_lds_b128 %0, %1, off"
               :: "v"(laddr), "v"(g)
               : "memory");
#endif
}

__device__ __forceinline__ void wait_async0() {
#if __has_builtin(__builtin_amdgcn_s_wait_asynccnt)
  __builtin_amdgcn_s_wait_asynccnt(0);
#else
  asm volatile("s_wait_asynccnt 0" ::: "memory");
#endif
}

__device__ __forceinline__ float rowsum(const float* q, int c0, int c1, int c2) {
  return q[c0] + 2.0f * q[c1] + q[c2];
}

__global__ __launch_bounds__(256) void blurpool3d_kernel(
    const float* __restrict__ x, float* __restrict__ out) {
  __shared__ float smem[NSLOT * PLANEF];   // 38,080 bytes

  const int tid = threadIdx.x;
  const int b   = blockIdx.x;
  const int oht = b % 7;            // which 8-row output tile
  const int nc  = b / 7;            // fused (n, c) index, 0..255
  const int oh0 = oht * OHT;

  const float* src = x   + (size_t)nc * (IDm * IHm * IWm);
  float*       dst = out + (size_t)nc * (ODm * OHm * OWm);

  // DMA one input plane (17 rows x 112 floats) into LDS slot `slot`,
  // clamping the top row for replicate padding. 476 B128 transfers.
  auto issue_plane = [&](int slot, int id) {
    float* lbase = &smem[slot * PLANEF];
    for (int i = tid; i < NVEC; i += 256) {
      int r = i / 28;                    // staged row 0..16
      int s = i % 28;                    // 16B chunk within row
      int ih = 2 * oh0 - 1 + r;
      if (ih < 0) ih = 0;                // replicate top edge (bottom never OOB)
      const float* g = src + ((size_t)id * IHm + ih) * IWm + s * 4;
      async_copy_b128(g, lbase + r * ROWF + s * 4);
    }
  };

  // Prologue: planes for od=0 are clamp(-1,0,1) = {0, 0, 1}.
  issue_plane(0, 0);
  issue_plane(1, 0);
  issue_plane(2, 1);
  wait_async0();
  __syncthreads();

  for (int od = 0; od < ODm; ++od) {
    // Prefetch the two planes needed by od+1 into the two free slots.
    // Those slots were last read before the previous barrier -> no WAR hazard.
    if (od < ODm - 1) {
      issue_plane((2 * od + 3) % NSLOT, 2 * od + 2);
      issue_plane((2 * od + 4) % NSLOT, 2 * od + 3);
    }

    const float* P0 = &smem[((2 * od    ) % NSLOT) * PLANEF]; // id = 2*od-1 (clamped)
    const float* P1 = &smem[((2 * od + 1) % NSLOT) * PLANEF]; // id = 2*od
    const float* P2 = &smem[((2 * od + 2) % NSLOT) * PLANEF]; // id = 2*od+1

    int o = tid;
#pragma unroll
    for (int it = 0; it < 2; ++it, o += 256) {   // 448 outputs / 256 threads
      if (o < OHT * OWm) {
        const int ohl = o / OWm;
        const int ow  = o - ohl * OWm;
        const int c1 = 2 * ow;
        const int c0 = (c1 > 0) ? c1 - 1 : 0;    // replicate left edge
        const int c2 = c1 + 1;                   // <= 111, never OOB
        const int r0 = 2 * ohl;

        const float* q0 = P0 + r0 * ROWF;
        const float* q1 = P1 + r0 * ROWF;
        const float* q2 = P2 + r0 * ROWF;

        float s0 =        rowsum(q0,            c0, c1, c2)
                 + 2.0f * rowsum(q0 + ROWF,     c0, c1, c2)
                 +        rowsum(q0 + 2 * ROWF, c0, c1, c2);
        float s1 =        rowsum(q1,            c0, c1, c2)
                 + 2.0f * rowsum(q1 + ROWF,     c0, c1, c2)
                 +        rowsum(q1 + 2 * ROWF, c0, c1, c2);
        float s2 =        rowsum(q2,            c0, c1, c2)
                 + 2.0f * rowsum(q2 + ROWF,     c0, c1, c2)
                 +        rowsum(q2 + 2 * ROWF, c0, c1, c2);

        float acc = (s0 + 2.0f * s1 + s2) * (1.0f / 64.0f);
        dst[((size_t)od * OHm + (oh0 + ohl)) * OWm + ow] = acc;
      }
    }

    wait_async0();      // next iteration's planes are resident
    __syncthreads();    // everyone done reading + loads visible
  }
}

extern "C" void kernel_launch(void* const* d_in, const int* in_sizes, int n_in,
                              void* d_out, int out_size, void* d_ws, size_t ws_size,
                              hipStream_t stream) {
  (void)in_sizes; (void)n_in; (void)out_size; (void)d_ws; (void)ws_size;
  const float* x = (const float*)d_in[0];
  float* out = (float*)d_out;
  // One block per (n*c, oh-tile): 256 * 7 = 1792 blocks, 256 threads (8 waves).
  blurpool3d_kernel<<<dim3(NB * CH * 7), dim3(256), 0, stream>>>(x, out);
}